// TOTEMVQVAE_90151363543330
// MI455X (gfx1250) — compile-verified
//
#include <hip/hip_runtime.h>
#include <hip/hip_bf16.h>

// TOTEM VQ-VAE forward for MI455X (gfx1250, wave32).
//   out layout: [x_recon 32*8*4096 | z_e 32*64*2049 | z_q 32*64*2049 | indices 32*2049]
// Roofline: ~80 MB total traffic @ 23.3 TB/s dominates; quantize GEMM (4.3 GFLOP)
// is mapped onto V_WMMA_F32_16X16X4_F32 (native f32 -> exact distances for argmin).

#define LQ   4096
#define LO   2049
#define NB   32
#define CIN  8
#define CLAT 64
#define KEMB 512
#define EDIM 64

typedef float v2f __attribute__((ext_vector_type(2)));
typedef float v8f __attribute__((ext_vector_type(8)));

// ---------------------------------------------------------------- encode ----
// z_e[n][co][lo] = b[co] + sum_{ci,k} x[n][ci][2*lo-2+k] * w[co][ci][k]
__global__ void vq_encode(const float* __restrict__ x,
                          const float* __restrict__ w,
                          const float* __restrict__ b,
                          float* __restrict__ ze) {
  __shared__ float wl[CLAT * CIN * 4];
  __shared__ float bl[CLAT];
  for (int i = threadIdx.x; i < CLAT * CIN * 4; i += blockDim.x) wl[i] = w[i];
  for (int i = threadIdx.x; i < CLAT; i += blockDim.x) bl[i] = b[i];
  __syncthreads();

  long gid = (long)blockIdx.x * blockDim.x + threadIdx.x;   // over N*CLAT*LO
  if (gid >= (long)NB * CLAT * LO) return;
  int  lo = (int)(gid % LO);
  long t  = gid / LO;
  int  co = (int)(t % CLAT);
  int  n  = (int)(t / CLAT);

  float acc = bl[co];
  const float* xn = x + (long)n * CIN * LQ;
  const float* wc = wl + co * (CIN * 4);
  int p0 = 2 * lo - 2;
#pragma unroll
  for (int ci = 0; ci < CIN; ++ci) {
#pragma unroll
    for (int k = 0; k < 4; ++k) {
      int p = p0 + k;
      float xv = (p >= 0 && p < LQ) ? xn[(long)ci * LQ + p] : 0.0f;
      acc = fmaf(xv, wc[ci * 4 + k], acc);
    }
  }
  ze[gid] = acc;
}

// -------------------------------------------------------------- quantize ----
// One wave handles 16 consecutive flat rows (g = n*LO + l).
// A = codebook tile (16 codes x 64), B = 16 z-rows (64 x 16), 16 K-steps of 4.
// dist_rel[m] = cb_sq[m] - 2*dot (z_sq is a per-row constant -> argmin-invariant).
#define CBPAD 66   // row stride in LDS floats: (r*66) % 64 = 2r -> conflict-free

__global__ void vq_quantize(const float* __restrict__ ze,
                            const float* __restrict__ cb,
                            float* __restrict__ zq,
                            float* __restrict__ ind) {
  extern __shared__ float lds[];
  float* cbl  = lds;                 // KEMB * CBPAD
  float* cbsq = lds + KEMB * CBPAD;  // KEMB

  for (int i = threadIdx.x; i < KEMB * EDIM; i += blockDim.x) {
    int m = i >> 6, c = i & 63;
    cbl[m * CBPAD + c] = cb[i];
  }
  for (int m = threadIdx.x; m < KEMB; m += blockDim.x) {
    float s = 0.0f;
    const float* row = cb + (long)m * EDIM;
#pragma unroll 8
    for (int c = 0; c < EDIM; ++c) s = fmaf(row[c], row[c], s);
    cbsq[m] = s;
  }
  __syncthreads();

  const int wave  = threadIdx.x >> 5;
  const int lane  = threadIdx.x & 31;
  const int group = blockIdx.x * (blockDim.x >> 5) + wave;
  const int ngroups = (NB * LO) / 16;                 // 65568/16 = 4098
  if (group >= ngroups) return;                       // wave-uniform exit

  const int r = lane & 15;          // z-row within group / B column
  const int h = lane >> 4;          // lane half -> K/M sub-offset
  const long g = (long)group * 16 + r;                // flat row id
  const int n = (int)(g / LO);
  const int l = (int)(g % LO);
  const float* zrow = ze + (long)n * CLAT * LO + l;   // + c*LO per channel

  // Preload B fragments: lane holds z[r][4*kk + 2h], z[r][4*kk + 2h + 1]
  v2f bfrag[16];
#pragma unroll
  for (int kk = 0; kk < 16; ++kk) {
    int c = 4 * kk + 2 * h;
    bfrag[kk].x = zrow[(long)c * LO];
    bfrag[kk].y = zrow[(long)(c + 1) * LO];
  }

  float minv = 3.402823466e38f;
  int   mini = 0;
  for (int t = 0; t < KEMB / 16; ++t) {               // 32 codebook tiles
    v8f acc = {};
    const float* arow = cbl + (t * 16 + r) * CBPAD + 2 * h;
#pragma unroll
    for (int kk = 0; kk < 16; ++kk) {
      v2f afrag;
      afrag.x = arow[4 * kk];
      afrag.y = arow[4 * kk + 1];
      acc = __builtin_amdgcn_wmma_f32_16x16x4_f32(
          false, afrag, false, bfrag[kk], (short)0, acc, false, false);
    }
    const int mbase = t * 16 + 8 * h;                 // C layout: VGPR v -> M=v(+8)
#pragma unroll
    for (int v = 0; v < 8; ++v) {
      int m = mbase + v;
      float d = cbsq[m] - 2.0f * acc[v];
      if (d < minv) { minv = d; mini = m; }           // codes ascend -> first-min
    }
  }
  // Merge lane halves (each saw a disjoint half of the 512 codes)
  float ov = __shfl_xor(minv, 16, 32);
  int   oi = __shfl_xor(mini, 16, 32);
  if (ov < minv || (ov == minv && oi < mini)) { minv = ov; mini = oi; }

  if (h == 0) ind[g] = (float)mini;

  // z_q[n][c][l] = codebook[mini][c]
  float* zqrow = zq + (long)n * CLAT * LO + l;
#pragma unroll
  for (int j = 0; j < 32; ++j) {
    int c = 2 * j + h;
    zqrow[(long)c * LO] = cbl[mini * CBPAD + c];
  }
}

// ---------------------------------------------------------------- decode ----
// ConvTranspose1d(stride=2,pad=2,k=4): for output t,
//   even t: x += w_t[ci][co][1]*zq[t/2]     + w_t[ci][co][3]*zq[t/2+1]
//   odd  t: x += w_t[ci][co][0]*zq[(t-1)/2] + w_t[ci][co][2]*zq[(t+1)/2]
// where w_t[ci][co][k] = tconv_w[co][ci][3-k]. All m indices stay in [0,2048].
__global__ void vq_decode(const float* __restrict__ zq,
                          const float* __restrict__ tw,
                          const float* __restrict__ tb,
                          float* __restrict__ xr) {
  __shared__ float wl[4][CLAT][CIN];   // wl[k][co][ci] = tw[co][ci][3-k]
  __shared__ float bl[CIN];
  for (int i = threadIdx.x; i < CLAT * CIN * 4; i += blockDim.x) {
    int co = i >> 5, rem = i & 31, ci = rem >> 2, k = rem & 3;
    wl[k][co][ci] = tw[(co * CIN + ci) * 4 + (3 - k)];
  }
  if (threadIdx.x < CIN) bl[threadIdx.x] = tb[threadIdx.x];
  __syncthreads();

  long gid = (long)blockIdx.x * blockDim.x + threadIdx.x;   // over N*LQ
  if (gid >= (long)NB * LQ) return;
  int t = (int)(gid % LQ);
  int n = (int)(gid / LQ);
  int par = t & 1;
  int m1  = par ? (t - 1) >> 1 : t >> 1;
  int kA  = par ? 0 : 1;
  int kB  = par ? 2 : 3;

  float acc[CIN];
#pragma unroll
  for (int ci = 0; ci < CIN; ++ci) acc[ci] = bl[ci];

  const float* zn = zq + (long)n * CLAT * LO + m1;
  for (int co = 0; co < CLAT; ++co) {
    float z0 = zn[(long)co * LO];
    float z1 = zn[(long)co * LO + 1];
#pragma unroll
    for (int ci = 0; ci < CIN; ++ci)
      acc[ci] = fmaf(wl[kA][co][ci], z0, fmaf(wl[kB][co][ci], z1, acc[ci]));
  }
  float* xo = xr + (long)n * CIN * LQ + t;
#pragma unroll
  for (int ci = 0; ci < CIN; ++ci) xo[(long)ci * LQ] = acc[ci];
}

// ---------------------------------------------------------------- launch ----
extern "C" void kernel_launch(void* const* d_in, const int* in_sizes, int n_in,
                              void* d_out, int out_size, void* d_ws, size_t ws_size,
                              hipStream_t stream) {
  const float* x        = (const float*)d_in[0];
  const float* conv_w   = (const float*)d_in[1];
  const float* conv_b   = (const float*)d_in[2];
  const float* codebook = (const float*)d_in[3];
  const float* tconv_w  = (const float*)d_in[4];
  const float* tconv_b  = (const float*)d_in[5];

  float* out = (float*)d_out;
  float* xr  = out;                                  // 32*8*4096   = 1048576
  float* ze  = xr + (long)NB * CIN * LQ;             // 32*64*2049  = 4196352
  float* zq  = ze + (long)NB * CLAT * LO;
  float* ind = zq + (long)NB * CLAT * LO;            // 32*2049     = 65568

  {  // encode: 4,196,352 outputs / 256 = 16392 blocks exactly
    long total = (long)NB * CLAT * LO;
    int blk = 256, grid = (int)((total + blk - 1) / blk);
    vq_encode<<<grid, blk, 0, stream>>>(x, conv_w, conv_b, ze);
  }
  {  // quantize: 4098 row-groups, 8 waves/block -> 513 blocks; 137 KB dynamic LDS
    int blk = 256;
    int ngroups = (NB * LO) / 16;
    int grid = (ngroups + (blk / 32) - 1) / (blk / 32);
    size_t ldsBytes = (size_t)(KEMB * CBPAD + KEMB) * sizeof(float);
    vq_quantize<<<grid, blk, ldsBytes, stream>>>(ze, codebook, zq, ind);
  }
  {  // decode: 131072 threads / 256 = 512 blocks exactly
    long total = (long)NB * LQ;
    int blk = 256, grid = (int)((total + blk - 1) / blk);
    vq_decode<<<grid, blk, 0, stream>>>(zq, tconv_w, tconv_b, xr);
  }
}